// KNNGraphCrossAttention_77884936946095
// MI455X (gfx1250) — compile-verified
//
#include <hip/hip_runtime.h>
#include <hip/hip_bf16.h>

typedef __attribute__((ext_vector_type(2))) float v2f;
typedef __attribute__((ext_vector_type(8))) float v8f;

constexpr int Bb   = 2;
constexpr int Nn   = 50000;
constexpr int Kk   = 16;
constexpr int Cc   = 256;
constexpr int ROWS = Bb * Nn;   // 100000, divisible by 16

// ---------------------------------------------------------------------------
// Pass 1: fused Q/K/V projections with fp32 WMMA (16x16x4).
//   out = X @ W^T + b   (then * sens_w row-scale for K and V)
// One wave (32 threads) computes a 16-row x 64-col strip (4 col tiles).
// A layout (16x4 f32): lane<16 -> row=lane, K={k,k+1}; lane>=16 -> row=lane-16, K={k+2,k+3}
// B layout (4x16 f32): mirrored, lane indexes the output column; B=W^T means a
// contiguous float2 from weight row (col).
// D layout: VGPR r -> row = r + 8*(lane>>4), col = lane&15.
// ---------------------------------------------------------------------------
__global__ __launch_bounds__(32) void proj3_wmma(
    const float* __restrict__ q_feat, const float* __restrict__ kv_feat,
    const float* __restrict__ Wq, const float* __restrict__ bq,
    const float* __restrict__ Wk, const float* __restrict__ bk,
    const float* __restrict__ Wv, const float* __restrict__ bv,
    const float* __restrict__ sens,
    float* __restrict__ Qo, float* __restrict__ Ko, float* __restrict__ Vo)
{
    const int which = blockIdx.z;                  // 0=Q, 1=K, 2=V
    const float* X    = (which == 0) ? q_feat : kv_feat;
    const float* W    = (which == 0) ? Wq : (which == 1 ? Wk : Wv);
    const float* bias = (which == 0) ? bq : (which == 1 ? bk : bv);
    float*       Out  = (which == 0) ? Qo : (which == 1 ? Ko : Vo);
    const bool scale  = (which != 0);

    const int lane  = threadIdx.x;       // 0..31
    const int half  = lane >> 4;         // 0/1: K sub-pair
    const int lr    = lane & 15;         // row (A) / col (B) within tile
    const int row0  = blockIdx.y * 16;
    const int colg0 = blockIdx.x * 64;   // 4 col tiles of 16

    const float* arow  = X + (size_t)(row0 + lr) * Cc + 2 * half;
    const float* wrow0 = W + (size_t)(colg0 + lr) * Cc + 2 * half;

    v8f acc0 = {}, acc1 = {}, acc2 = {}, acc3 = {};
    #pragma unroll 4
    for (int k = 0; k < Cc; k += 4) {
        v2f a  = *(const v2f*)(arow + k);
        v2f b0 = *(const v2f*)(wrow0 + k);
        v2f b1 = *(const v2f*)(wrow0 + 16 * Cc + k);
        v2f b2 = *(const v2f*)(wrow0 + 32 * Cc + k);
        v2f b3 = *(const v2f*)(wrow0 + 48 * Cc + k);
        acc0 = __builtin_amdgcn_wmma_f32_16x16x4_f32(false, a, false, b0, (short)0, acc0, false, false);
        acc1 = __builtin_amdgcn_wmma_f32_16x16x4_f32(false, a, false, b1, (short)0, acc1, false, false);
        acc2 = __builtin_amdgcn_wmma_f32_16x16x4_f32(false, a, false, b2, (short)0, acc2, false, false);
        acc3 = __builtin_amdgcn_wmma_f32_16x16x4_f32(false, a, false, b3, (short)0, acc3, false, false);
    }

    // per-lane row scales (sens_w applies per token n = global row % N)
    float sv[8];
    #pragma unroll
    for (int r = 0; r < 8; ++r) {
        int g = row0 + half * 8 + r;
        float s = 1.0f;
        if (scale) { int n = g - ((g >= Nn) ? Nn : 0); s = sens[n]; }
        sv[r] = s;
    }

    auto emit = [&](v8f acc, int t) {
        const int col = colg0 + t * 16 + lr;
        const float bc = bias[col];
        #pragma unroll
        for (int r = 0; r < 8; ++r) {
            int g = row0 + half * 8 + r;
            Out[(size_t)g * Cc + col] = (acc[r] + bc) * sv[r];
        }
    };
    emit(acc0, 0); emit(acc1, 1); emit(acc2, 2); emit(acc3, 3);
}

// ---------------------------------------------------------------------------
// Pass 2: per-query KNN attention + softmax + residual + LayerNorm.
// One wave32 per query; lane owns 8 channels. Dots reduce via 5 butterfly
// shuffles; softmax over K=16 replicated per lane; LN = 2 more reductions.
// ---------------------------------------------------------------------------
__global__ __launch_bounds__(256) void knn_attn(
    const float* __restrict__ Q, const float* __restrict__ Kw,
    const float* __restrict__ Vw, const int* __restrict__ knn,
    const float* __restrict__ ln_g, const float* __restrict__ ln_b,
    float* __restrict__ out)
{
    const int gtid = blockIdx.x * blockDim.x + threadIdx.x;
    const int wave = gtid >> 5;
    const int lane = gtid & 31;
    if (wave >= ROWS) return;

    const int b = (wave >= Nn) ? 1 : 0;
    const int n = wave - b * Nn;
    const int coff = lane * 8;

    const float* qrow = Q + (size_t)wave * Cc + coff;
    float4 q01 = *(const float4*)(qrow);
    float4 q23 = *(const float4*)(qrow + 4);
    float q[8] = { q01.x, q01.y, q01.z, q01.w, q23.x, q23.y, q23.z, q23.w };

    const int* knr = knn + (size_t)n * Kk;   // wave-uniform -> scalar loads
    int idxs[Kk];
    #pragma unroll
    for (int k = 0; k < Kk; ++k) idxs[k] = knr[k];

    const float* Kb = Kw + (size_t)b * Nn * Cc;
    const float* Vb = Vw + (size_t)b * Nn * Cc;

    float a[Kk];
    #pragma unroll
    for (int k = 0; k < Kk; ++k) {
        const float* kr = Kb + (size_t)idxs[k] * Cc + coff;
        float4 k01 = *(const float4*)(kr);
        float4 k23 = *(const float4*)(kr + 4);
        float s = 0.0f;
        s = fmaf(q[0], k01.x, s); s = fmaf(q[1], k01.y, s);
        s = fmaf(q[2], k01.z, s); s = fmaf(q[3], k01.w, s);
        s = fmaf(q[4], k23.x, s); s = fmaf(q[5], k23.y, s);
        s = fmaf(q[6], k23.z, s); s = fmaf(q[7], k23.w, s);
        #pragma unroll
        for (int off = 1; off < 32; off <<= 1) s += __shfl_xor(s, off, 32);
        a[k] = s * 0.0625f;                  // 1/sqrt(256)
    }

    // softmax over K (replicated per lane)
    float m = a[0];
    #pragma unroll
    for (int k = 1; k < Kk; ++k) m = fmaxf(m, a[k]);
    float ssum = 0.0f;
    #pragma unroll
    for (int k = 0; k < Kk; ++k) { a[k] = __expf(a[k] - m); ssum += a[k]; }
    const float inv = 1.0f / ssum;

    // out = attn @ V + Q (residual folded in)
    float o[8];
    #pragma unroll
    for (int i = 0; i < 8; ++i) o[i] = q[i];
    #pragma unroll
    for (int k = 0; k < Kk; ++k) {
        const float p = a[k] * inv;
        const float* vr = Vb + (size_t)idxs[k] * Cc + coff;
        float4 v01 = *(const float4*)(vr);
        float4 v23 = *(const float4*)(vr + 4);
        o[0] = fmaf(p, v01.x, o[0]); o[1] = fmaf(p, v01.y, o[1]);
        o[2] = fmaf(p, v01.z, o[2]); o[3] = fmaf(p, v01.w, o[3]);
        o[4] = fmaf(p, v23.x, o[4]); o[5] = fmaf(p, v23.y, o[5]);
        o[6] = fmaf(p, v23.z, o[6]); o[7] = fmaf(p, v23.w, o[7]);
    }

    // LayerNorm across C=256
    float s1 = 0.0f;
    #pragma unroll
    for (int i = 0; i < 8; ++i) s1 += o[i];
    #pragma unroll
    for (int off = 1; off < 32; off <<= 1) s1 += __shfl_xor(s1, off, 32);
    const float mu = s1 * (1.0f / Cc);
    float s2 = 0.0f;
    #pragma unroll
    for (int i = 0; i < 8; ++i) { float d = o[i] - mu; s2 = fmaf(d, d, s2); }
    #pragma unroll
    for (int off = 1; off < 32; off <<= 1) s2 += __shfl_xor(s2, off, 32);
    const float rstd = rsqrtf(s2 * (1.0f / Cc) + 1e-5f);

    float4 g01 = *(const float4*)(ln_g + coff);
    float4 g23 = *(const float4*)(ln_g + coff + 4);
    float4 b01 = *(const float4*)(ln_b + coff);
    float4 b23 = *(const float4*)(ln_b + coff + 4);
    float4 r0, r1;
    r0.x = (o[0] - mu) * rstd * g01.x + b01.x;
    r0.y = (o[1] - mu) * rstd * g01.y + b01.y;
    r0.z = (o[2] - mu) * rstd * g01.z + b01.z;
    r0.w = (o[3] - mu) * rstd * g01.w + b01.w;
    r1.x = (o[4] - mu) * rstd * g23.x + b23.x;
    r1.y = (o[5] - mu) * rstd * g23.y + b23.y;
    r1.z = (o[6] - mu) * rstd * g23.z + b23.z;
    r1.w = (o[7] - mu) * rstd * g23.w + b23.w;

    float* orow = out + (size_t)wave * Cc + coff;
    *(float4*)(orow)     = r0;
    *(float4*)(orow + 4) = r1;
}

extern "C" void kernel_launch(void* const* d_in, const int* in_sizes, int n_in,
                              void* d_out, int out_size, void* d_ws, size_t ws_size,
                              hipStream_t stream) {
    const float* q_feat  = (const float*)d_in[0];
    const float* kv_feat = (const float*)d_in[1];
    const int*   knn_idx = (const int*)  d_in[2];
    const float* sens_w  = (const float*)d_in[3];
    const float* Wq_w    = (const float*)d_in[4];
    const float* Wq_b    = (const float*)d_in[5];
    const float* Wk_w    = (const float*)d_in[6];
    const float* Wk_b    = (const float*)d_in[7];
    const float* Wv_w    = (const float*)d_in[8];
    const float* Wv_b    = (const float*)d_in[9];
    const float* ln_g    = (const float*)d_in[10];
    const float* ln_b    = (const float*)d_in[11];
    float* out = (float*)d_out;

    // workspace: Q | K*w | V*w, each B*N*C floats (3 * 102.4 MB)
    const size_t planeElems = (size_t)ROWS * Cc;
    float* Qbuf = (float*)d_ws;
    float* Kbuf = Qbuf + planeElems;
    float* Vbuf = Kbuf + planeElems;

    // Pass 1: projections. grid = (colGroups=4, rowTiles=6250, which=3)
    dim3 g1(Cc / 64, ROWS / 16, 3);
    proj3_wmma<<<g1, 32, 0, stream>>>(q_feat, kv_feat,
                                      Wq_w, Wq_b, Wk_w, Wk_b, Wv_w, Wv_b,
                                      sens_w, Qbuf, Kbuf, Vbuf);

    // Pass 2: one wave per query, 8 waves per block.
    const int threads = 256;
    const int waves_per_block = threads / 32;
    const int blocks = (ROWS + waves_per_block - 1) / waves_per_block;
    knn_attn<<<blocks, threads, 0, stream>>>(Qbuf, Kbuf, Vbuf, knn_idx,
                                             ln_g, ln_b, out);
}